// SAGE_40913858462098
// MI455X (gfx1250) — compile-verified
//
#include <hip/hip_runtime.h>
#include <hip/hip_bf16.h>

typedef float v2f __attribute__((ext_vector_type(2)));
typedef float v8f __attribute__((ext_vector_type(8)));

#ifndef THREADS
#define THREADS 256
#endif

// ---------------------------------------------------------------------------
// Zero-fill fp32 workspace (agg buffers + degree) — grid-stride.
// ---------------------------------------------------------------------------
__global__ __launch_bounds__(THREADS) void sage_zero_f32(float* __restrict__ p, size_t n) {
    size_t i      = (size_t)blockIdx.x * blockDim.x + threadIdx.x;
    size_t stride = (size_t)gridDim.x * blockDim.x;
    for (; i < n; i += stride) p[i] = 0.0f;
}

// ---------------------------------------------------------------------------
// Edge scatter: agg[dst] += feats[src] (64 feats/edge), deg[dst] += 1.
// 16 threads per edge, float4 gather + 4 global_atomic_add_f32 each.
// All feature matrices fit in the 192MB L2, so these atomics and the random
// row gathers are L2-resident.
// ---------------------------------------------------------------------------
__global__ __launch_bounds__(THREADS) void sage_scatter_edges(
        const float* __restrict__ feats,   // [N, 64]
        const int*   __restrict__ src,     // [E]
        const int*   __restrict__ dst,     // [E]
        float*       __restrict__ agg,     // [N, 64]
        float*       __restrict__ deg,     // [N] or nullptr
        int n_edges)
{
    long long gid = (long long)blockIdx.x * blockDim.x + threadIdx.x;
    int e = (int)(gid >> 4);
    if (e >= n_edges) return;
    int part = (int)(gid & 15);           // which 4-float chunk of the 64 feats

    int s = src[e];
    int d = dst[e];

    const float4 v = *(const float4*)(feats + (size_t)s * 64 + part * 4);
    float* ap = agg + (size_t)d * 64 + part * 4;
    atomicAdd(ap + 0, v.x);
    atomicAdd(ap + 1, v.y);
    atomicAdd(ap + 2, v.z);
    atomicAdd(ap + 3, v.w);
    if (deg != nullptr && part == 0) atomicAdd(deg + d, 1.0f);
}

// ---------------------------------------------------------------------------
// Fused SAGE layer GEMM using V_WMMA_F32_16X16X4_F32 (fp32 WMMA, wave32):
//   OUT[16 rows] = act( X@W_self + (AGG * 1/max(deg,1)) @ W_neigh + bias )
// One wave per 16-row tile; 8 waves (256 threads) per block; tile guard is
// wave-uniform so EXEC is all-1s at every WMMA.
//
// fp32 A 16x4 layout (ISA 7.12.2): lane L (L<16) row m=L holds K=k,k+1 in
// v[0],v[1]; lane L+16 holds K=k+2,k+3. B 4x16 mirrors C/D row striping:
// lane gives column n=L%16, v[0]/v[1] give K rows (k | k+2) / (k+1 | k+3).
// C/D 16x16: element (M = r + 8*(lane>=16), N = lane%16) in vgpr r.
// ---------------------------------------------------------------------------
template <int NOUT, bool RELU>
__global__ __launch_bounds__(THREADS) void sage_gemm(
        const float* __restrict__ X,       // [N, 64]
        const float* __restrict__ AGG,     // [N, 64]
        const float* __restrict__ DEG,     // [N]
        const float* __restrict__ Wself,   // [64, NOUT]
        const float* __restrict__ Wneigh,  // [64, NOUT]
        const float* __restrict__ bias,    // [NOUT]
        float*       __restrict__ OUT,     // [N, NOUT]
        int n_nodes)
{
    constexpr int CT = NOUT / 16;          // output column tiles
    const int wave  = threadIdx.x >> 5;
    const int lane  = threadIdx.x & 31;
    const int tile  = blockIdx.x * (THREADS / 32) + wave;
    const int row0  = tile * 16;
    if (row0 >= n_nodes) return;           // wave-uniform: EXEC stays full

    const int m     = lane & 15;           // A row within tile / B,C column
    const int khalf = lane >> 4;           // 0: K=k,k+1   1: K=k+2,k+3
    int rowi = row0 + m;
    if (rowi >= n_nodes) rowi = n_nodes - 1;   // clamp (stores guarded below)

    const float dv  = DEG[rowi];
    const float inv = 1.0f / fmaxf(dv, 1.0f);

    const float* __restrict__ xrow = X   + (size_t)rowi * 64;
    const float* __restrict__ arow = AGG + (size_t)rowi * 64;

    v8f acc[CT];
#pragma unroll
    for (int ct = 0; ct < CT; ++ct) acc[ct] = (v8f){};

#pragma unroll
    for (int k = 0; k < 64; k += 4) {
        const int kk = k + khalf * 2;
        v2f a_s = *(const v2f*)(xrow + kk);           // 8B aligned (kk even)
        v2f a_r = *(const v2f*)(arow + kk);
        v2f a_n = { a_r.x * inv, a_r.y * inv };       // mean-agg row scaling
#pragma unroll
        for (int ct = 0; ct < CT; ++ct) {
            const int n = ct * 16 + m;
            v2f b_s = { Wself[(size_t)kk * NOUT + n],
                        Wself[(size_t)(kk + 1) * NOUT + n] };
            acc[ct] = __builtin_amdgcn_wmma_f32_16x16x4_f32(
                          false, a_s, false, b_s, (short)0, acc[ct], false, false);
            v2f b_n = { Wneigh[(size_t)kk * NOUT + n],
                        Wneigh[(size_t)(kk + 1) * NOUT + n] };
            acc[ct] = __builtin_amdgcn_wmma_f32_16x16x4_f32(
                          false, a_n, false, b_n, (short)0, acc[ct], false, false);
        }
    }

#pragma unroll
    for (int ct = 0; ct < CT; ++ct) {
        const int n  = ct * 16 + m;
        const float bb = bias[n];
#pragma unroll
        for (int r = 0; r < 8; ++r) {
            const int M = r + 8 * khalf;
            if (row0 + M < n_nodes) {
                float v = acc[ct][r] + bb;
                if (RELU) v = fmaxf(v, 0.0f);
                OUT[(size_t)(row0 + M) * NOUT + n] = v;
            }
        }
    }
}

// ---------------------------------------------------------------------------
// Driver
// ---------------------------------------------------------------------------
extern "C" void kernel_launch(void* const* d_in, const int* in_sizes, int n_in,
                              void* d_out, int out_size, void* d_ws, size_t ws_size,
                              hipStream_t stream) {
    const float* x   = (const float*)d_in[0];
    const int*   src = (const int*)d_in[1];
    const int*   dst = (const int*)d_in[2];
    const float* Ws1 = (const float*)d_in[3];
    const float* Wn1 = (const float*)d_in[4];
    const float* b1  = (const float*)d_in[5];
    const float* Ws2 = (const float*)d_in[6];
    const float* Wn2 = (const float*)d_in[7];
    const float* b2  = (const float*)d_in[8];
    float* out = (float*)d_out;

    const int n_nodes = in_sizes[0] / 64;
    const int n_edges = in_sizes[1];

    // workspace carve: [agg1 | agg2 | deg | h]
    float* agg1 = (float*)d_ws;
    float* agg2 = agg1 + (size_t)n_nodes * 64;
    float* deg  = agg2 + (size_t)n_nodes * 64;
    float* h    = deg  + (size_t)n_nodes;

    const size_t zero_elems = (size_t)n_nodes * 128 + (size_t)n_nodes; // agg1+agg2+deg
    {
        int blocks = (int)((zero_elems + THREADS - 1) / THREADS);
        if (blocks > 65535) blocks = 65535;
        hipLaunchKernelGGL(sage_zero_f32, dim3(blocks), dim3(THREADS), 0, stream,
                           agg1, zero_elems);
    }

    const long long scat_threads = (long long)n_edges * 16;
    const int scat_blocks = (int)((scat_threads + THREADS - 1) / THREADS);

    const int tiles       = (n_nodes + 15) / 16;
    const int gemm_blocks = (tiles + (THREADS / 32) - 1) / (THREADS / 32);

    // ---- Layer 1 ----
    hipLaunchKernelGGL(sage_scatter_edges, dim3(scat_blocks), dim3(THREADS), 0, stream,
                       x, src, dst, agg1, deg, n_edges);
    hipLaunchKernelGGL((sage_gemm<64, true>), dim3(gemm_blocks), dim3(THREADS), 0, stream,
                       x, agg1, deg, Ws1, Wn1, b1, h, n_nodes);

    // ---- Layer 2 ----
    hipLaunchKernelGGL(sage_scatter_edges, dim3(scat_blocks), dim3(THREADS), 0, stream,
                       h, src, dst, agg2, (float*)nullptr, n_edges);
    hipLaunchKernelGGL((sage_gemm<32, false>), dim3(gemm_blocks), dim3(THREADS), 0, stream,
                       h, agg2, deg, Ws2, Wn2, b2, out, n_nodes);
}